// EstimatorQNN_65481071405578
// MI455X (gfx1250) — compile-verified
//
#include <hip/hip_runtime.h>
#include <hip/hip_bf16.h>

typedef __attribute__((ext_vector_type(2))) float v2f;
typedef __attribute__((ext_vector_type(8))) float v8f;

// ---------------------------------------------------------------------------
// Quanvolution: one thread = one 2x2 patch -> 4-qubit statevector in registers
// Flat index i = a*8 + b*4 + c*2 + d  (qubit0 -> bit3 ... qubit3 -> bit0)
// ---------------------------------------------------------------------------

template <int M>
__device__ __forceinline__ void rot_x(float (&re)[16], float (&im)[16], float cg, float sg) {
#pragma unroll
  for (int i = 0; i < 16; ++i) {
    if (i & M) continue;
    const int j = i | M;
    float r0 = re[i], i0 = im[i], r1 = re[j], i1 = im[j];
    // U = [[c, -i s], [-i s, c]]
    re[i] = cg * r0 + sg * i1;
    im[i] = cg * i0 - sg * r1;
    re[j] = sg * i0 + cg * r1;
    im[j] = -sg * r0 + cg * i1;
  }
}

template <int M>
__device__ __forceinline__ void rot_y(float (&re)[16], float (&im)[16], float cg, float sg) {
#pragma unroll
  for (int i = 0; i < 16; ++i) {
    if (i & M) continue;
    const int j = i | M;
    float r0 = re[i], i0 = im[i], r1 = re[j], i1 = im[j];
    // U = [[c, -s], [s, c]]
    re[i] = cg * r0 - sg * r1;
    im[i] = cg * i0 - sg * i1;
    re[j] = sg * r0 + cg * r1;
    im[j] = sg * i0 + cg * i1;
  }
}

template <int M>
__device__ __forceinline__ void rot_z(float (&re)[16], float (&im)[16], float cg, float sg) {
#pragma unroll
  for (int i = 0; i < 16; ++i) {
    float r = re[i], ii = im[i];
    if (!(i & M)) {        // * (c - i s)
      re[i] = cg * r + sg * ii;
      im[i] = cg * ii - sg * r;
    } else {               // * (c + i s)
      re[i] = cg * r - sg * ii;
      im[i] = cg * ii + sg * r;
    }
  }
}

template <int MC, int MT>
__device__ __forceinline__ void cnot(float (&re)[16], float (&im)[16]) {
#pragma unroll
  for (int i = 0; i < 16; ++i) {
    if ((i & MC) && !(i & MT)) {
      const int j = i | MT;
      float tr = re[i]; re[i] = re[j]; re[j] = tr;
      float ti = im[i]; im[i] = im[j]; im[j] = ti;
    }
  }
}

__global__ void quanv_kernel(const float* __restrict__ x,
                             const float* __restrict__ qp,
                             float* __restrict__ feats) {
  const int t = blockIdx.x * blockDim.x + threadIdx.x;
  if (t >= 4096 * 196) return;
  const int n  = t / 196;
  const int p  = t - n * 196;
  const int pi = p / 14;
  const int pj = p - pi * 14;

  const float* img = x + (size_t)n * 784;
  const int base = (2 * pi) * 28 + 2 * pj;
  const float t0 = img[base], t1 = img[base + 1];
  const float t2 = img[base + 28], t3 = img[base + 29];

  float c0, s0, c1, s1, c2, s2, c3, s3;
  __sincosf(0.5f * t0, &s0, &c0);
  __sincosf(0.5f * t1, &s1, &c1);
  __sincosf(0.5f * t2, &s2, &c2);
  __sincosf(0.5f * t3, &s3, &c3);
  const float w0[2] = {c0, s0}, w1[2] = {c1, s1}, w2v[2] = {c2, s2}, w3[2] = {c3, s3};

  float re[16], im[16];
#pragma unroll
  for (int a = 0; a < 2; ++a)
#pragma unroll
    for (int b = 0; b < 2; ++b)
#pragma unroll
      for (int c = 0; c < 2; ++c)
#pragma unroll
        for (int d = 0; d < 2; ++d) {
          re[a * 8 + b * 4 + c * 2 + d] = w0[a] * w1[b] * w2v[c] * w3[d];
          im[a * 8 + b * 4 + c * 2 + d] = 0.0f;
        }

  // gate params (uniform across wave)
  float gc0, gs0, gc1, gs1, gc2, gs2, gc3, gs3, gc4, gs4;
  __sincosf(0.5f * qp[0], &gs0, &gc0);
  __sincosf(0.5f * qp[1], &gs1, &gc1);
  __sincosf(0.5f * qp[2], &gs2, &gc2);
  __sincosf(0.5f * qp[3], &gs3, &gc3);
  __sincosf(0.5f * qp[4], &gs4, &gc4);

  rot_x<8>(re, im, gc0, gs0);   // RX(q0)
  rot_y<4>(re, im, gc1, gs1);   // RY(q1)
  rot_z<2>(re, im, gc2, gs2);   // RZ(q2)
  cnot<8, 4>(re, im);           // CNOT(0,1)
  rot_x<1>(re, im, gc3, gs3);   // RX(q3)
  cnot<2, 1>(re, im);           // CNOT(2,3)
  rot_y<8>(re, im, gc4, gs4);   // RY(q0)
  cnot<4, 2>(re, im);           // CNOT(1,2)

  float e0 = 0.f, e1 = 0.f, e2 = 0.f, e3 = 0.f;
#pragma unroll
  for (int i = 0; i < 16; ++i) {
    const float pr = re[i] * re[i] + im[i] * im[i];
    e0 += (i & 8) ? -pr : pr;
    e1 += (i & 4) ? -pr : pr;
    e2 += (i & 2) ? -pr : pr;
    e3 += (i & 1) ? -pr : pr;
  }
  *(float4*)(feats + (size_t)n * 784 + p * 4) = make_float4(e0, e1, e2, e3);
}

// ---------------------------------------------------------------------------
// GEMM1: hidden(4096x128) = relu(feats(4096x784) @ W1^T(784x128) + b1)
// One block (8 waves) per 16-row tile; A tile staged to LDS via async copy
// (GLOBAL_LOAD_ASYNC_TO_LDS_B128, ASYNCcnt), WMMAs fed from LDS; B from L2.
// LDS row stride padded to 788 floats (788 % 64 = 20 -> conflict-free b64).
// ---------------------------------------------------------------------------
#define LDS_STRIDE 788

__global__ void gemm1_kernel(const float* __restrict__ A,    // feats, 4096 x 784
                             const float* __restrict__ W1,   // 128 x 784
                             const float* __restrict__ b1,
                             float* __restrict__ H) {        // 4096 x 128
  __shared__ float aTile[16 * LDS_STRIDE];

  const int mTile = blockIdx.x;        // 256 row tiles
  const int tid   = threadIdx.x;
  const int lane  = tid & 31;
  const int nTile = tid >> 5;          // 8 waves -> 8 col tiles

  // ---- stage A tile (16 x 784 floats) into LDS with async b128 copies ----
  {
    const float* aBase = A + mTile * (16 * 784);
#pragma unroll
    for (int pass = 0; pass < 13; ++pass) {      // 16*196 = 3136 float4 chunks
      const int idx = pass * 256 + tid;
      if (idx < 16 * 196) {
        const int row = idx / 196;
        const int c4  = idx - row * 196;
        const float* g = aBase + row * 784 + c4 * 4;
        // low 32 bits of a generic __shared__ pointer == wave-local LDS addr
        const unsigned ldsAddr =
            (unsigned)(size_t)(const void*)(aTile + row * LDS_STRIDE + c4 * 4);
        asm volatile("global_load_async_to_lds_b128 %0, %1, off"
                     :: "v"(ldsAddr), "v"(g) : "memory");
      }
    }
    asm volatile("s_wait_asynccnt 0x0" ::: "memory");
  }
  __syncthreads();

  // ---- WMMA main loop: A from LDS, B from global (L2-resident W1) ----
  const int khalf = (lane >> 4) << 1;             // K-pair base: 0 or 2
  const int col   = (nTile << 4) + (lane & 15);   // output col for this lane
  const float* aLds = aTile + (lane & 15) * LDS_STRIDE + khalf;
  const float* bPtr = W1 + col * 784 + khalf;     // W1 row-major == B^T

  v8f acc = {};
#pragma unroll 4
  for (int k = 0; k < 784; k += 4) {
    v2f a = *(const v2f*)(aLds + k);
    v2f b = *(const v2f*)(bPtr + k);
    acc = __builtin_amdgcn_wmma_f32_16x16x4_f32(
        /*neg_a=*/false, a, /*neg_b=*/false, b,
        /*c_mod=*/(short)0, acc, /*reuse_a=*/false, /*reuse_b=*/false);
  }

  const float bias    = b1[col];
  const int   rowBase = (mTile << 4) + ((lane >> 4) << 3);  // C/D: row = r + 8*(lane/16)
#pragma unroll
  for (int r = 0; r < 8; ++r) {
    float v = acc[r] + bias;
    H[(rowBase + r) * 128 + col] = v > 0.0f ? v : 0.0f;
  }
}

// ---------------------------------------------------------------------------
// GEMM2: logits(4096x10) = hidden(4096x128) @ W2^T(128x10) + b2, N padded to 16
// ---------------------------------------------------------------------------
__global__ void gemm2_kernel(const float* __restrict__ H,    // 4096 x 128
                             const float* __restrict__ W2,   // 10 x 128
                             const float* __restrict__ b2,
                             float* __restrict__ out) {      // 4096 x 10
  const int wave  = (blockIdx.x * blockDim.x + threadIdx.x) >> 5;
  const int lane  = threadIdx.x & 31;
  const int mTile = wave;                          // 256 row tiles

  const int row   = (mTile << 4) + (lane & 15);
  const int j     = lane & 15;                     // padded output col
  const int khalf = (lane >> 4) << 1;

  const float msk = (j < 10) ? 1.0f : 0.0f;        // zero-mask (keep EXEC full)
  const int   jc  = (j < 10) ? j : 0;

  const float* aPtr = H  + row * 128 + khalf;
  const float* bPtr = W2 + jc * 128 + khalf;

  v8f acc = {};
#pragma unroll
  for (int k = 0; k < 128; k += 4) {
    v2f a = *(const v2f*)(aPtr + k);
    v2f b = *(const v2f*)(bPtr + k);
    b.x *= msk;
    b.y *= msk;
    acc = __builtin_amdgcn_wmma_f32_16x16x4_f32(
        false, a, false, b, (short)0, acc, false, false);
  }

  if (j < 10) {
    const float bias    = b2[j];
    const int   rowBase = (mTile << 4) + ((lane >> 4) << 3);
#pragma unroll
    for (int r = 0; r < 8; ++r)
      out[(rowBase + r) * 10 + j] = acc[r] + bias;
  }
}

// ---------------------------------------------------------------------------
extern "C" void kernel_launch(void* const* d_in, const int* in_sizes, int n_in,
                              void* d_out, int out_size, void* d_ws, size_t ws_size,
                              hipStream_t stream) {
  const float* x  = (const float*)d_in[0];   // (4096,1,28,28)
  const float* qp = (const float*)d_in[1];   // (5,)
  const float* W1 = (const float*)d_in[2];   // (128,784)
  const float* b1 = (const float*)d_in[3];   // (128,)
  const float* W2 = (const float*)d_in[4];   // (10,128)
  const float* b2 = (const float*)d_in[5];   // (10,)
  float* out = (float*)d_out;                // (4096,10)

  float* feats  = (float*)d_ws;              // 4096*784 floats (12.85 MB)
  float* hidden = feats + 4096 * 784;        // 4096*128 floats (2.10 MB)

  // 1) quanvolution: 802816 patches, 1 thread each
  quanv_kernel<<<(4096 * 196) / 256, 256, 0, stream>>>(x, qp, feats);

  // 2) GEMM1 + bias + relu: 256 blocks, one 16-row tile each, 8 waves x 8 col tiles
  gemm1_kernel<<<256, 256, 0, stream>>>(feats, W1, b1, hidden);

  // 3) GEMM2 + bias: 256 tiles, 8 waves/block
  gemm2_kernel<<<32, 256, 0, stream>>>(hidden, W2, b2, out);
}